// FoldIteration_36429912604699
// MI455X (gfx1250) — compile-verified
//
#include <hip/hip_runtime.h>
#include <math.h>

typedef float v2f __attribute__((ext_vector_type(2)));
typedef float v8f __attribute__((ext_vector_type(8)));

#define kN 1024
#define kH 12

__device__ __forceinline__ v8f wmma4(v2f a, v2f b, v8f c) {
  // D = A(16x4 f32) * B(4x16 f32) + C(16x16 f32)
  return __builtin_amdgcn_wmma_f32_16x16x4_f32(false, a, false, b, (short)0, c, false, false);
}

#define SCALAR_W 0.14433756729740643f   /* sqrt(1/48)  */
#define POINT_W  0.13608276348795434f   /* sqrt(1/54)  */
#define A2D_W    0.5773502691896258f    /* sqrt(1/3)   */

// ---------------- Kernel 1: fused input projections ------------------------
// proj[i][o], o: [0,192) q | [192,576) kv | [576,720) qp | [720,1152) kvp
__global__ __launch_bounds__(256) void k_proj(
    const float* __restrict__ x1d,
    const float* __restrict__ Wq, const float* __restrict__ bq,
    const float* __restrict__ Wkv, const float* __restrict__ bkv,
    const float* __restrict__ Wqp, const float* __restrict__ bqp,
    const float* __restrict__ Wkvp, const float* __restrict__ bkvp,
    float* __restrict__ proj) {
  __shared__ __align__(16) float Abuf[16 * 388];  // padded stride: no bank conflicts
  const int i0 = blockIdx.x * 16;
  for (int t = threadIdx.x; t < 16 * 384; t += 256) {
    int r = t / 384, c = t % 384;
    Abuf[r * 388 + c] = x1d[(size_t)(i0 + r) * 384 + c];
  }
  __syncthreads();
  const int wave = threadIdx.x >> 5, lane = threadIdx.x & 31;
  const int lo = lane & 15, hi = lane >> 4;
  for (int nt = wave; nt < 72; nt += 8) {
    int o0 = nt * 16;
    const float* W; const float* B; int wo;
    if (o0 < 192)      { W = Wq;   B = bq;   wo = o0; }
    else if (o0 < 576) { W = Wkv;  B = bkv;  wo = o0 - 192; }
    else if (o0 < 720) { W = Wqp;  B = bqp;  wo = o0 - 576; }
    else               { W = Wkvp; B = bkvp; wo = o0 - 720; }
    const float* wrow = W + (size_t)(wo + lo) * 384;
    v8f acc = {0.f,0.f,0.f,0.f,0.f,0.f,0.f,0.f};
    for (int kb = 0; kb < 96; ++kb) {
      int k0 = 4 * kb + 2 * hi;
      v2f a; a.x = Abuf[lo * 388 + k0]; a.y = Abuf[lo * 388 + k0 + 1];
      v2f b = *(const v2f*)(wrow + k0);
      acc = wmma4(a, b, acc);
    }
    float bias = B[wo + lo];
#pragma unroll
    for (int r = 0; r < 8; ++r) {
      int m = r + 8 * hi;
      proj[(size_t)(i0 + m) * 1152 + o0 + lo] = acc[r] + bias;
    }
  }
}

// ---------------- Kernel 2: build extended Q~/K~ (K=32) and Vcat -----------
__global__ __launch_bounds__(256) void k_build(
    const float* __restrict__ proj, const float* __restrict__ rot,
    const float* __restrict__ trans, const float* __restrict__ tpw,
    float* __restrict__ Qt, float* __restrict__ Kt, float* __restrict__ Vcat) {
  int gid = blockIdx.x * blockDim.x + threadIdx.x;
  if (gid >= kN * kH) return;
  int i = gid / kH, h = gid % kH;
  const float* pr = proj + (size_t)i * 1152;
  float R[9], T[3];
#pragma unroll
  for (int t = 0; t < 9; ++t) R[t] = rot[i * 9 + t];
#pragma unroll
  for (int c = 0; c < 3; ++c) T[c] = trans[i * 3 + c];
  float x = tpw[h];
  float sp = (x > 20.f) ? x : log1pf(__expf(x));
  float pw = sp * POINT_W;

  float qpg[3][4];
  {
    float ql[3][4];
    for (int c = 0; c < 3; ++c)
      for (int p = 0; p < 4; ++p) ql[c][p] = pr[576 + c * 48 + h * 4 + p];
    for (int c = 0; c < 3; ++c)
      for (int p = 0; p < 4; ++p)
        qpg[c][p] = R[c*3+0]*ql[0][p] + R[c*3+1]*ql[1][p] + R[c*3+2]*ql[2][p] + T[c];
  }
  float kvl[3][12], kvg[3][12];
  for (int c = 0; c < 3; ++c)
    for (int d = 0; d < 12; ++d) kvl[c][d] = pr[720 + c * 144 + h * 12 + d];
  for (int c = 0; c < 3; ++c)
    for (int d = 0; d < 12; ++d)
      kvg[c][d] = R[c*3+0]*kvl[0][d] + R[c*3+1]*kvl[1][d] + R[c*3+2]*kvl[2][d] + T[c];

  float qn = 0.f, kn = 0.f;
  for (int c = 0; c < 3; ++c)
    for (int p = 0; p < 4; ++p) { qn += qpg[c][p]*qpg[c][p]; kn += kvg[c][p]*kvg[c][p]; }

  float* q = Qt + ((size_t)h * kN + i) * 32;
  float* k = Kt + ((size_t)h * kN + i) * 32;
  float* v = Vcat + ((size_t)h * kN + i) * 48;
  for (int d = 0; d < 16; ++d) q[d] = SCALAR_W * pr[h * 16 + d];
  for (int c = 0; c < 3; ++c)
    for (int p = 0; p < 4; ++p) q[16 + c * 4 + p] = pw * qpg[c][p];
  q[28] = -0.5f * pw * qn; q[29] = 1.f; q[30] = 0.f; q[31] = 0.f;

  for (int d = 0; d < 16; ++d) k[d] = pr[192 + h * 32 + d];
  for (int c = 0; c < 3; ++c)
    for (int p = 0; p < 4; ++p) k[16 + c * 4 + p] = kvg[c][p];
  k[28] = 1.f; k[29] = -0.5f * pw * kn; k[30] = 0.f; k[31] = 0.f;

  for (int d = 0; d < 16; ++d) v[d] = pr[192 + h * 32 + 16 + d];
  for (int c = 0; c < 3; ++c)
    for (int p = 0; p < 8; ++p) v[16 + c * 8 + p] = kvg[c][4 + p];
  for (int d = 40; d < 48; ++d) v[d] = 0.f;
}

// Pre-swizzle W2d^T into WMMA-B operand order: [kb(32)][lane(32)] x float2
__global__ void k_w2dswz(const float* __restrict__ W2d, float* __restrict__ swz) {
  int t = blockIdx.x * blockDim.x + threadIdx.x;
  if (t >= 1024) return;
  int kb = t / 32, l = t % 32, lo = l & 15, hi = l >> 4;
  int c0 = 4 * kb + 2 * hi;
  float x = 0.f, y = 0.f;
  if (lo < kH) { x = W2d[lo * 128 + c0]; y = W2d[lo * 128 + c0 + 1]; }
  swz[2 * t] = x; swz[2 * t + 1] = y;
}

// ---------------- Kernel 3: logits = Q~.K~ + A2D_W*a2d + mask (537MB pass) -
__global__ __launch_bounds__(256) void k_logits(
    const float* __restrict__ x2d, const float* __restrict__ w2dswz,
    const float* __restrict__ b2d, const float* __restrict__ mask,
    const float* __restrict__ Qt, const float* __restrict__ Kt,
    float* __restrict__ attn) {
  __shared__ __align__(16) float b2lds[2048];
  __shared__ __align__(16) float a2dlds[16 * 16 * 17];  // [i_loc][j_loc][h] pad17
  const int j0 = blockIdx.x * 16, i0 = blockIdx.y * 16;
  for (int t = threadIdx.x; t < 2048; t += 256) b2lds[t] = w2dswz[t];
  __syncthreads();
  const int wave = threadIdx.x >> 5, lane = threadIdx.x & 31;
  const int lo = lane & 15, hi = lane >> 4;

  // phase A: a2d for all heads, rows = j_loc, cols = h, K = 128
  // x2d is streamed exactly once -> non-temporal: keep attn/Qt/Kt hot in L2
  for (int il = wave; il < 16; il += 8) {
    const float* arow = x2d + ((size_t)(i0 + il) * kN + (j0 + lo)) * 128;
    v8f acc = {0.f,0.f,0.f,0.f,0.f,0.f,0.f,0.f};
    for (int kb = 0; kb < 32; ++kb) {
      int k0 = 4 * kb + 2 * hi;
      v2f a = __builtin_nontemporal_load((const v2f*)(arow + k0));
      v2f b = *(const v2f*)(b2lds + (kb * 32 + lane) * 2);
      acc = wmma4(a, b, acc);
    }
#pragma unroll
    for (int r = 0; r < 8; ++r) {
      int m = r + 8 * hi;                      // j_loc
      a2dlds[(il * 16 + m) * 17 + lo] = acc[r]; // col = head
    }
  }
  __syncthreads();

  // phase B: per-head 16x16 logits tile, K = 32
  for (int h = wave; h < kH; h += 8) {
    const float* qrow = Qt + ((size_t)h * kN + i0 + lo) * 32;
    const float* krow = Kt + ((size_t)h * kN + j0 + lo) * 32;
    v8f acc = {0.f,0.f,0.f,0.f,0.f,0.f,0.f,0.f};
#pragma unroll
    for (int kb = 0; kb < 8; ++kb) {
      int k0 = 4 * kb + 2 * hi;
      v2f a = *(const v2f*)(qrow + k0);
      v2f b = *(const v2f*)(krow + k0);
      acc = wmma4(a, b, acc);
    }
    float bh = b2d[h];
    float mj = mask[j0 + lo];
#pragma unroll
    for (int r = 0; r < 8; ++r) {
      int m = r + 8 * hi;                      // i_loc
      float mi = mask[i0 + m];
      float val = acc[r] + A2D_W * (a2dlds[(m * 16 + lo) * 17 + h] + bh)
                  - 100000.f * (1.f - mi * mj);
      attn[((size_t)h * kN + i0 + m) * kN + j0 + lo] = val;
    }
  }
}

// ---------------- Kernel 4: row softmax (one wave32 per row) ---------------
__global__ __launch_bounds__(256) void k_softmax(float* __restrict__ attn) {
  int row = blockIdx.x * 8 + (threadIdx.x >> 5);
  int lane = threadIdx.x & 31;
  float* p = attn + (size_t)row * kN;
  float v[32];
  float mx = -INFINITY;
#pragma unroll
  for (int t = 0; t < 32; ++t) { v[t] = p[lane + 32 * t]; mx = fmaxf(mx, v[t]); }
  for (int s = 16; s > 0; s >>= 1) mx = fmaxf(mx, __shfl_xor(mx, s, 32));
  float sum = 0.f;
#pragma unroll
  for (int t = 0; t < 32; ++t) { v[t] = __expf(v[t] - mx); sum += v[t]; }
  for (int s = 16; s > 0; s >>= 1) sum += __shfl_xor(sum, s, 32);
  float inv = 1.f / sum;
#pragma unroll
  for (int t = 0; t < 32; ++t) p[lane + 32 * t] = v[t] * inv;
}

// ---------------- Kernel 5: attn @ Vcat (per head, N=48, K=1024) -----------
__global__ __launch_bounds__(256) void k_av(
    const float* __restrict__ attn, const float* __restrict__ Vcat,
    float* __restrict__ svout) {
  __shared__ __align__(16) float red[8 * 768];
  const int h = blockIdx.y, i0 = blockIdx.x * 16;
  const int wave = threadIdx.x >> 5, lane = threadIdx.x & 31;
  const int lo = lane & 15, hi = lane >> 4;
  const float* arow = attn + ((size_t)h * kN + i0 + lo) * kN;
  const float* vb = Vcat + (size_t)h * kN * 48;
  v8f acc[3];
#pragma unroll
  for (int t = 0; t < 3; ++t) acc[t] = (v8f){0.f,0.f,0.f,0.f,0.f,0.f,0.f,0.f};
  for (int kb = wave * 32; kb < wave * 32 + 32; ++kb) {
    int k0 = 4 * kb + 2 * hi;
    v2f a = *(const v2f*)(arow + k0);
#pragma unroll
    for (int t = 0; t < 3; ++t) {
      v2f b;
      b.x = vb[(size_t)k0 * 48 + t * 16 + lo];
      b.y = vb[(size_t)(k0 + 1) * 48 + t * 16 + lo];
      acc[t] = wmma4(a, b, acc[t]);
    }
  }
#pragma unroll
  for (int t = 0; t < 3; ++t)
#pragma unroll
    for (int r = 0; r < 8; ++r)
      red[wave * 768 + t * 256 + (r + 8 * hi) * 16 + lo] = acc[t][r];
  __syncthreads();
  for (int o = threadIdx.x; o < 768; o += 256) {
    float s = 0.f;
#pragma unroll
    for (int w = 0; w < 8; ++w) s += red[w * 768 + o];
    int t = o >> 8, rem = o & 255, m = rem >> 4, n = rem & 15;
    svout[((size_t)h * kN + i0 + m) * 48 + t * 16 + n] = s;
  }
}

// ---------------- Kernel 6: r2d (second 537MB pass), M=heads, N=128 --------
__global__ __launch_bounds__(256) void k_r2d(
    const float* __restrict__ attn, const float* __restrict__ x2d,
    float* __restrict__ F) {
  __shared__ __align__(16) float red[4 * 2048];  // 32 KB, two-phase reduce
  const int i = blockIdx.x;
  const int wave = threadIdx.x >> 5, lane = threadIdx.x & 31;
  const int lo = lane & 15, hi = lane >> 4;
  const int hh = lo < kH ? lo : kH - 1;         // pad rows read valid memory
  const float* arow = attn + ((size_t)hh * kN + i) * kN;
  const float* xrow = x2d + (size_t)i * kN * 128;
  v8f acc[8];
#pragma unroll
  for (int t = 0; t < 8; ++t) acc[t] = (v8f){0.f,0.f,0.f,0.f,0.f,0.f,0.f,0.f};
  for (int kb = wave * 32; kb < wave * 32 + 32; ++kb) {
    int k0 = 4 * kb + 2 * hi;
    v2f a = *(const v2f*)(arow + k0);
#pragma unroll
    for (int t = 0; t < 8; ++t) {
      v2f b;  // x2d streamed once -> non-temporal, protect attn in L2
      b.x = __builtin_nontemporal_load(xrow + (size_t)k0 * 128 + t * 16 + lo);
      b.y = __builtin_nontemporal_load(xrow + (size_t)(k0 + 1) * 128 + t * 16 + lo);
      acc[t] = wmma4(a, b, acc[t]);
    }
  }
  if (wave < 4) {
#pragma unroll
    for (int t = 0; t < 8; ++t)
#pragma unroll
      for (int r = 0; r < 8; ++r)
        red[wave * 2048 + t * 256 + (r + 8 * hi) * 16 + lo] = acc[t][r];
  }
  __syncthreads();
  if (wave >= 4) {
#pragma unroll
    for (int t = 0; t < 8; ++t)
#pragma unroll
      for (int r = 0; r < 8; ++r)
        red[(wave - 4) * 2048 + t * 256 + (r + 8 * hi) * 16 + lo] += acc[t][r];
  }
  __syncthreads();
  for (int o = threadIdx.x; o < 2048; o += 256) {
    float s = red[o] + red[2048 + o] + red[4096 + o] + red[6144 + o];
    int t = o >> 8, rem = o & 255, m = rem >> 4, n = rem & 15;
    if (m < kH) F[(size_t)i * 2112 + 576 + m * 128 + t * 16 + n] = s;
  }
}

// ---------------- Kernel 7: epilogue features (scalar copy + local frame) --
__global__ __launch_bounds__(192) void k_epi(
    const float* __restrict__ svout, const float* __restrict__ rot,
    const float* __restrict__ trans, float* __restrict__ F) {
  const int i = blockIdx.x;
  const int t = threadIdx.x;
  if (t < 192) {
    int h = t >> 4, d = t & 15;
    F[(size_t)i * 2112 + t] = svout[((size_t)h * kN + i) * 48 + d];
  }
  if (t < 96) {
    int h = t >> 3, p = t & 7;
    float g[3];
#pragma unroll
    for (int c = 0; c < 3; ++c)
      g[c] = svout[((size_t)h * kN + i) * 48 + 16 + c * 8 + p] - trans[i * 3 + c];
    float l[3];
#pragma unroll
    for (int d = 0; d < 3; ++d)  // rot^T: l[d] = sum_j rot[j][d]*g[j]
      l[d] = rot[i*9 + 0*3 + d]*g[0] + rot[i*9 + 1*3 + d]*g[1] + rot[i*9 + 2*3 + d]*g[2];
    float pn = sqrtf(1e-8f + l[0]*l[0] + l[1]*l[1] + l[2]*l[2]);
    F[(size_t)i * 2112 + 192 + h * 8 + p] = l[0];
    F[(size_t)i * 2112 + 288 + h * 8 + p] = l[1];
    F[(size_t)i * 2112 + 384 + h * 8 + p] = l[2];
    F[(size_t)i * 2112 + 480 + h * 8 + p] = pn;
  }
}

// ---------------- Kernel 8: final projection (1024x2112)@(2112x384)+b ------
__global__ __launch_bounds__(256) void k_final(
    const float* __restrict__ F, const float* __restrict__ Wout,
    const float* __restrict__ bout, float* __restrict__ out) {
  __shared__ __align__(16) float red[8 * 256];
  const int i0 = blockIdx.x * 16, o0 = blockIdx.y * 16;
  const int wave = threadIdx.x >> 5, lane = threadIdx.x & 31;
  const int lo = lane & 15, hi = lane >> 4;
  const float* arow = F + (size_t)(i0 + lo) * 2112;
  const float* brow = Wout + (size_t)(o0 + lo) * 2112;
  v8f acc = {0.f,0.f,0.f,0.f,0.f,0.f,0.f,0.f};
  for (int kb = wave * 66; kb < wave * 66 + 66; ++kb) {
    int k0 = 4 * kb + 2 * hi;
    v2f a = *(const v2f*)(arow + k0);
    v2f b = *(const v2f*)(brow + k0);
    acc = wmma4(a, b, acc);
  }
#pragma unroll
  for (int r = 0; r < 8; ++r)
    red[wave * 256 + (r + 8 * hi) * 16 + lo] = acc[r];
  __syncthreads();
  for (int o = threadIdx.x; o < 256; o += 256) {
    float s = 0.f;
#pragma unroll
    for (int w = 0; w < 8; ++w) s += red[w * 256 + o];
    int m = o >> 4, n = o & 15;
    out[(size_t)(i0 + m) * 384 + o0 + n] = s + bout[o0 + n];
  }
}

extern "C" void kernel_launch(void* const* d_in, const int* in_sizes, int n_in,
                              void* d_out, int out_size, void* d_ws, size_t ws_size,
                              hipStream_t stream) {
  (void)in_sizes; (void)n_in; (void)out_size; (void)ws_size;
  const float* x1d   = (const float*)d_in[0];
  const float* x2d   = (const float*)d_in[1];
  const float* mask  = (const float*)d_in[2];
  const float* rot   = (const float*)d_in[3];
  const float* trans = (const float*)d_in[4];
  const float* Wq    = (const float*)d_in[5];
  const float* bq    = (const float*)d_in[6];
  const float* Wkv   = (const float*)d_in[7];
  const float* bkv   = (const float*)d_in[8];
  const float* Wqp   = (const float*)d_in[9];
  const float* bqp   = (const float*)d_in[10];
  const float* Wkvp  = (const float*)d_in[11];
  const float* bkvp  = (const float*)d_in[12];
  const float* tpw   = (const float*)d_in[13];
  const float* W2d   = (const float*)d_in[14];
  const float* b2d   = (const float*)d_in[15];
  const float* Wout  = (const float*)d_in[16];
  const float* bout  = (const float*)d_in[17];
  float* out = (float*)d_out;

  float* ws     = (float*)d_ws;
  float* proj   = ws;                    // 1024*1152      = 1,179,648
  float* Qt     = proj   + 1179648;      // 12*1024*32     =   393,216
  float* Kt     = Qt     + 393216;       //                =   393,216
  float* Vcat   = Kt     + 393216;       // 12*1024*48     =   589,824
  float* w2dswz = Vcat   + 589824;       //                =     2,048
  float* svout  = w2dswz + 2048;         // 12*1024*48     =   589,824
  float* Fbuf   = svout  + 589824;       // 1024*2112      = 2,162,688
  float* attn   = Fbuf   + 2162688;      // 12*1024*1024   = 12,582,912
  // total ~68.3 MB

  k_proj<<<64, 256, 0, stream>>>(x1d, Wq, bq, Wkv, bkv, Wqp, bqp, Wkvp, bkvp, proj);
  k_build<<<48, 256, 0, stream>>>(proj, rot, trans, tpw, Qt, Kt, Vcat);
  k_w2dswz<<<4, 256, 0, stream>>>(W2d, w2dswz);
  k_logits<<<dim3(64, 64), 256, 0, stream>>>(x2d, w2dswz, b2d, mask, Qt, Kt, attn);
  k_softmax<<<1536, 256, 0, stream>>>(attn);
  k_av<<<dim3(64, kH), 256, 0, stream>>>(attn, Vcat, svout);
  k_r2d<<<1024, 256, 0, stream>>>(attn, x2d, Fbuf);
  k_epi<<<1024, 192, 0, stream>>>(svout, rot, trans, Fbuf);
  k_final<<<dim3(64, 24), 256, 0, stream>>>(Fbuf, Wout, bout, out);
}